// ApproachNetIndex_37752762532105
// MI455X (gfx1250) — compile-verified
//
#include <hip/hip_runtime.h>
#include <hip/hip_bf16.h>

typedef __attribute__((ext_vector_type(2))) float v2f;
typedef __attribute__((ext_vector_type(8))) float v8f;

#define BB 8
#define NN 4096
#define CC 256
#define VV 300
#define NPNP 20000
#define KK 2048
#define PP 2

// ---------------------------------------------------------------------------
// Small helper kernels
// ---------------------------------------------------------------------------
__global__ void pad_w_kernel(const float* __restrict__ src, float* __restrict__ dst,
                             int rows, int cols, int Rp, int Cp) {
    int i = blockIdx.x * blockDim.x + threadIdx.x;
    if (i >= Rp * Cp) return;
    int r = i / Cp, c = i % Cp;
    dst[i] = (r < rows && c < cols) ? src[r * cols + c] : 0.0f;
}

__global__ void pad_vec_kernel(const float* __restrict__ src, float* __restrict__ dst,
                               int n, int np) {
    int i = blockIdx.x * blockDim.x + threadIdx.x;
    if (i < np) dst[i] = (i < n) ? src[i] : 0.0f;
}

__global__ void gen_templates_kernel(float* __restrict__ tmpl) {
    int i = blockIdx.x * blockDim.x + threadIdx.x;
    if (i >= VV) return;
    const double PI = 3.14159265358979323846;
    double phi = (sqrt(5.0) - 1.0) / 2.0;
    double z = (2.0 * (double)i + 1.0) / (double)VV - 1.0;
    double rad = sqrt(fmax(1.0 - z * z, 0.0));
    double ang = 2.0 * PI * (double)i * phi;
    tmpl[i * 3 + 0] = (float)(rad * cos(ang));
    tmpl[i * 3 + 1] = (float)(rad * sin(ang));
    tmpl[i * 3 + 2] = (float)z;
}

// Stable argsort(1 - mask): masked seed positions first (in order), then rest.
__global__ void build_idx_kernel(const int* __restrict__ index_mask,
                                 const int* __restrict__ fp2_inds,
                                 int* __restrict__ idxArr) {
    int b = blockIdx.x;
    if (threadIdx.x != 0) return;
    int cnt = 0;
    for (int n = 0; n < NN && cnt < KK; ++n)
        if (index_mask[(size_t)b * NPNP + fp2_inds[b * NN + n]] != 0)
            idxArr[b * KK + cnt++] = n;
    for (int n = 0; n < NN && cnt < KK; ++n)
        if (index_mask[(size_t)b * NPNP + fp2_inds[b * NN + n]] == 0)
            idxArr[b * KK + cnt++] = n;
}

// ---------------------------------------------------------------------------
// WMMA f32 GEMM: Y[b,m,n] = sum_k W[m,k] * g(X[b,k,n]) + bias[m]
// NORM: g(x) = relu(x*scale[k] + shift[k]) (fused BN+ReLU of previous layer),
// else identity. Specialized at compile time -> branch-free inner loop.
// Workgroup: 256 threads = 8 waves (2 M x 4 N), tile 64(M) x 128(N), K step 4.
// MODE 0: write Y (padded activation buffer, layout b*Mp*N + m*N + n)
// MODE 1: final layer split: m<2 -> objectness (B,2,N); 2<=m<realM ->
//         view_score transposed (B,N,300).
// ---------------------------------------------------------------------------
template <bool NORM, int MODE>
__global__ __launch_bounds__(256)
void gemm_wmma_kernel(const float* __restrict__ W, const float* __restrict__ X,
                      const float* __restrict__ bias,
                      const float* __restrict__ scale, const float* __restrict__ shift,
                      float* __restrict__ Y, float* __restrict__ out_obj,
                      float* __restrict__ out_view,
                      int Mp, int Kp, int realM) {
    const int b = blockIdx.z;
    const int tile_n0 = blockIdx.x * 128;
    const int tile_m0 = blockIdx.y * 64;
    const int tid  = threadIdx.x;
    const int wave = tid >> 5;
    const int lane = tid & 31;
    const int mbase = tile_m0 + (wave >> 2) * 32;
    const int nbase = tile_n0 + (wave & 3) * 32;
    const int lm  = lane & 15;          // row/col-in-tile for A/B frags
    const int lk2 = (lane >> 4) << 1;   // K pair base: 0 or 2

    const float* Xb = X + (size_t)b * Kp * NN;
    const float* arow0 = W + (size_t)(mbase +      lm) * Kp + lk2;
    const float* arow1 = W + (size_t)(mbase + 16 + lm) * Kp + lk2;
    const float* xcol0 = Xb + (size_t)lk2 * NN + nbase + lm;
    const float* xcol1 = xcol0 + 16;

    v8f acc[2][2];
    const v8f vzero = {0.f, 0.f, 0.f, 0.f, 0.f, 0.f, 0.f, 0.f};
    acc[0][0] = vzero; acc[0][1] = vzero; acc[1][0] = vzero; acc[1][1] = vzero;

    for (int k0 = 0; k0 < Kp; k0 += 4) {
        // A fragments: lane l -> row lm, K = k0 + lk2 + {0,1} (contiguous pair)
        v2f a0 = *(const v2f*)(arow0 + k0);
        v2f a1 = *(const v2f*)(arow1 + k0);

        const int kk0 = k0 + lk2;
        float x00 = xcol0[(size_t)k0 * NN];
        float x01 = xcol0[(size_t)(k0 + 1) * NN];
        float x10 = xcol1[(size_t)k0 * NN];
        float x11 = xcol1[(size_t)(k0 + 1) * NN];
        if (NORM) {
            float s0 = scale[kk0], t0 = shift[kk0];
            float s1 = scale[kk0 + 1], t1 = shift[kk0 + 1];
            x00 = fmaxf(fmaf(x00, s0, t0), 0.0f);
            x01 = fmaxf(fmaf(x01, s1, t1), 0.0f);
            x10 = fmaxf(fmaf(x10, s0, t0), 0.0f);
            x11 = fmaxf(fmaf(x11, s1, t1), 0.0f);
        }
        v2f b0; b0.x = x00; b0.y = x01;
        v2f b1; b1.x = x10; b1.y = x11;

        acc[0][0] = __builtin_amdgcn_wmma_f32_16x16x4_f32(false, a0, false, b0,
                        (short)0, acc[0][0], false, false);
        acc[0][1] = __builtin_amdgcn_wmma_f32_16x16x4_f32(false, a0, false, b1,
                        (short)0, acc[0][1], false, false);
        acc[1][0] = __builtin_amdgcn_wmma_f32_16x16x4_f32(false, a1, false, b0,
                        (short)0, acc[1][0], false, false);
        acc[1][1] = __builtin_amdgcn_wmma_f32_16x16x4_f32(false, a1, false, b1,
                        (short)0, acc[1][1], false, false);
    }

    // Epilogue. C/D layout: VGPR r, lanes 0-15 -> M=r, N=lane; lanes 16-31 ->
    // M=r+8, N=lane-16.
    const int mhi = (lane >> 4) << 3;
    const int nlo = lane & 15;
    for (int mi = 0; mi < 2; ++mi) {
        for (int ni = 0; ni < 2; ++ni) {
            for (int r = 0; r < 8; ++r) {
                int m = mbase + mi * 16 + mhi + r;
                int n = nbase + ni * 16 + nlo;
                float y = acc[mi][ni][r] + bias[m];
                if (MODE == 0) {
                    Y[((size_t)b * Mp + m) * NN + n] = y;
                } else {
                    if (m < 2) {
                        out_obj[((size_t)b * 2 + m) * NN + n] = y;
                    } else if (m < realM) {
                        out_view[((size_t)b * NN + n) * VV + (m - 2)] = y;
                    }
                }
            }
        }
    }
}

// ---------------------------------------------------------------------------
// Per-channel mean/var over (B, N); produce fused BN scale/shift.
// ---------------------------------------------------------------------------
__global__ __launch_bounds__(256)
void stats_kernel(const float* __restrict__ Y, const float* __restrict__ gamma,
                  const float* __restrict__ beta, float* __restrict__ scaleArr,
                  float* __restrict__ shiftArr, int Mp, int realM) {
    const int m = blockIdx.x;
    __shared__ float ssum[256];
    __shared__ float ssq[256];
    float s = 0.0f, q = 0.0f;
    for (int i = threadIdx.x; i < BB * NN; i += 256) {
        int b = i >> 12, n = i & (NN - 1);
        float v = Y[((size_t)b * Mp + m) * NN + n];
        s += v; q += v * v;
    }
    ssum[threadIdx.x] = s; ssq[threadIdx.x] = q;
    __syncthreads();
    for (int off = 128; off > 0; off >>= 1) {
        if (threadIdx.x < off) {
            ssum[threadIdx.x] += ssum[threadIdx.x + off];
            ssq[threadIdx.x]  += ssq[threadIdx.x + off];
        }
        __syncthreads();
    }
    if (threadIdx.x == 0) {
        const float inv = 1.0f / (float)(BB * NN);
        float mean = ssum[0] * inv;
        float var  = ssq[0] * inv - mean * mean;
        float sc = 0.0f, sh = 0.0f;
        if (m < realM) {
            sc = gamma[m] * rsqrtf(var + 1e-5f);
            sh = beta[m] - mean * sc;
        }
        scaleArr[m] = sc;
        shiftArr[m] = sh;
    }
}

// ---------------------------------------------------------------------------
// Top-2 view scores per masked seed. One wave per seed; tie-break matches
// stable argsort(-score): score desc, view index asc.
// ---------------------------------------------------------------------------
__device__ __forceinline__ bool score_better(float a, int ia, float b, int ib) {
    return (a > b) || (a == b && ia < ib);
}

__global__ __launch_bounds__(256)
void top2_kernel(const float* __restrict__ view_score, const int* __restrict__ idxArr,
                 float* __restrict__ ss, int* __restrict__ si) {
    const int wave = threadIdx.x >> 5;
    const int lane = threadIdx.x & 31;
    const int gj = blockIdx.x * 8 + wave;      // [0, B*K)
    const int b = gj / KK, j = gj % KK;
    const int row = idxArr[b * KK + j];
    const float* vs = view_score + ((size_t)b * NN + row) * VV;

    float s1 = -3.402823466e38f; int i1 = 0x7fffffff;
    float s2 = -3.402823466e38f; int i2 = 0x7fffffff;
    for (int v = lane; v < VV; v += 32) {
        float s = vs[v];
        if (score_better(s, v, s1, i1)) { s2 = s1; i2 = i1; s1 = s; i1 = v; }
        else if (score_better(s, v, s2, i2)) { s2 = s; i2 = v; }
    }
    for (int off = 16; off > 0; off >>= 1) {
        float os1 = __shfl_xor(s1, off, 32); int oi1 = __shfl_xor(i1, off, 32);
        float os2 = __shfl_xor(s2, off, 32); int oi2 = __shfl_xor(i2, off, 32);
        if (score_better(os1, oi1, s1, i1)) {
            float ts = s1; int ti = i1;
            s1 = os1; i1 = oi1;
            if (score_better(ts, ti, os2, oi2)) { s2 = ts; i2 = ti; }
            else { s2 = os2; i2 = oi2; }
        } else {
            if (score_better(os1, oi1, s2, i2)) { s2 = os1; i2 = oi1; }
        }
    }
    if (lane == 0) {
        size_t base = (size_t)b * NN + 2 * j;
        ss[base] = s1;     si[base] = i1;
        ss[base + 1] = s2; si[base + 1] = i2;
    }
}

// ---------------------------------------------------------------------------
// Full descending sort (== top_k(ss, N)) per batch: bitonic sort of 4096
// (score, position) pairs in LDS. Tie-break: position asc (top_k stability).
// ---------------------------------------------------------------------------
__global__ __launch_bounds__(512)
void sort_kernel(const float* __restrict__ ss, float* __restrict__ skey,
                 int* __restrict__ spos) {
    const int b = blockIdx.x;
    __shared__ float s[NN];
    __shared__ int   p[NN];
    for (int i = threadIdx.x; i < NN; i += 512) {
        s[i] = ss[(size_t)b * NN + i];
        p[i] = i;
    }
    __syncthreads();
    for (int k = 2; k <= NN; k <<= 1) {
        for (int j = k >> 1; j > 0; j >>= 1) {
            for (int i = threadIdx.x; i < NN; i += 512) {
                int ix = i ^ j;
                if (ix > i) {
                    float a = s[i], c = s[ix];
                    int pa = p[i], pc = p[ix];
                    bool aFirst = score_better(a, pa, c, pc); // "a first" (desc)
                    bool descRegion = ((i & k) == 0);
                    bool doSwap = descRegion ? (!aFirst) : aFirst;
                    if (doSwap) {
                        s[i] = c; s[ix] = a;
                        p[i] = pc; p[ix] = pa;
                    }
                }
            }
            __syncthreads();
        }
    }
    for (int i = threadIdx.x; i < NN; i += 512) {
        skey[(size_t)b * NN + i] = s[i];
        spos[(size_t)b * NN + i] = p[i];
    }
}

// ---------------------------------------------------------------------------
// Final gathers + viewpoint_to_matrix.
// ---------------------------------------------------------------------------
__global__ __launch_bounds__(256)
void finalize_kernel(const float* __restrict__ skey, const int* __restrict__ spos,
                     const int* __restrict__ si, const int* __restrict__ idxArr,
                     const float* __restrict__ fp2_xyz, const float* __restrict__ tmpl,
                     float* __restrict__ tvs, float* __restrict__ tvi,
                     float* __restrict__ fin, float* __restrict__ fxyz,
                     float* __restrict__ vxyz, float* __restrict__ vrot) {
    int g = blockIdx.x * blockDim.x + threadIdx.x;
    if (g >= BB * NN) return;
    int b = g / NN;
    int pos = spos[g];
    int view = si[(size_t)b * NN + pos];
    int seed = idxArr[b * KK + (pos >> 1)];

    tvs[g] = skey[g];
    tvi[g] = (float)view;
    fin[g] = (float)seed;

    // fp2_xyz_new uses batch 0's xyz for all rows (matches reference).
    fxyz[g * 3 + 0] = fp2_xyz[seed * 3 + 0];
    fxyz[g * 3 + 1] = fp2_xyz[seed * 3 + 1];
    fxyz[g * 3 + 2] = fp2_xyz[seed * 3 + 2];

    float vx = tmpl[view * 3 + 0];
    float vy = tmpl[view * 3 + 1];
    float vz = tmpl[view * 3 + 2];
    vxyz[g * 3 + 0] = vx; vxyz[g * 3 + 1] = vy; vxyz[g * 3 + 2] = vz;

    // towards = -vp_xyz
    float axx = -vx, axy = -vy, axz = -vz;
    float ayx = -axy, ayy = axx, ayz = 0.0f;
    float ny = sqrtf(ayx * ayx + ayy * ayy);
    if (ny == 0.0f) { ayx = 0.0f; ayy = 1.0f; ayz = 0.0f; }
    float nx = sqrtf(axx * axx + axy * axy + axz * axz);
    axx /= nx; axy /= nx; axz /= nx;
    float nya = sqrtf(ayx * ayx + ayy * ayy + ayz * ayz);
    ayx /= nya; ayy /= nya; ayz /= nya;
    float azx = axy * ayz - axz * ayy;
    float azy = axz * ayx - axx * ayz;
    float azz = axx * ayy - axy * ayx;
    // stack([ax, ay, az], axis=-1): columns are ax, ay, az.
    float* R = vrot + (size_t)g * 9;
    R[0] = axx; R[1] = ayx; R[2] = azx;
    R[3] = axy; R[4] = ayy; R[5] = azy;
    R[6] = axz; R[7] = ayz; R[8] = azz;
}

// ---------------------------------------------------------------------------
// Launcher
// ---------------------------------------------------------------------------
extern "C" void kernel_launch(void* const* d_in, const int* in_sizes, int n_in,
                              void* d_out, int out_size, void* d_ws, size_t ws_size,
                              hipStream_t stream) {
    (void)in_sizes; (void)n_in; (void)out_size; (void)ws_size;
    // Inputs (setup_inputs order)
    const float* seed_features = (const float*)d_in[1];   // (B,C,N)
    const float* fp2_xyz       = (const float*)d_in[2];   // (B,N,3)
    const int*   fp2_inds      = (const int*)d_in[3];     // (B,N)
    const int*   index_mask    = (const int*)d_in[4];     // (B,NP)
    const float* W1    = (const float*)d_in[5];
    const float* b1    = (const float*)d_in[6];
    const float* g1    = (const float*)d_in[7];
    const float* beta1 = (const float*)d_in[8];
    const float* W2    = (const float*)d_in[9];
    const float* b2    = (const float*)d_in[10];
    const float* g2    = (const float*)d_in[11];
    const float* beta2 = (const float*)d_in[12];
    const float* W3    = (const float*)d_in[13];
    const float* b3    = (const float*)d_in[14];

    // Output layout (concatenated flat, float32)
    float* out = (float*)d_out;
    float* out_obj  = out;                                  // (B,2,N)      65536
    float* out_view = out + 65536;                          // (B,N,300)  9830400
    float* out_tvs  = out + 9895936;                        // (B,N)        32768
    float* out_tvi  = out + 9928704;                        // (B,N)        32768
    float* out_fin  = out + 9961472;                        // (B,N)        32768
    float* out_fxyz = out + 9994240;                        // (1,B*N,3)    98304
    float* out_vxyz = out + 10092544;                       // (B,N,3)      98304
    float* out_vrot = out + 10190848;                       // (B,N,3,3)   294912

    // Workspace carve-up (floats)
    float* ws = (float*)d_ws;
    size_t off = 0;
    const int M2P = 320, M3P = 320;            // padded output channels (mult of 64)
    float* W2p    = ws + off; off += (size_t)M2P * CC;     // 320x256
    float* W3p    = ws + off; off += (size_t)M3P * M2P;    // 320x320
    float* b2p    = ws + off; off += M2P;
    float* b3p    = ws + off; off += M3P;
    float* scale1 = ws + off; off += CC;
    float* shift1 = ws + off; off += CC;
    float* scale2 = ws + off; off += M2P;
    float* shift2 = ws + off; off += M2P;
    float* tmpl   = ws + off; off += 904;                  // 300x3 (+pad)
    float* Y1     = ws + off; off += (size_t)BB * CC * NN;   // 8x256x4096
    float* Y2     = ws + off; off += (size_t)BB * M2P * NN;  // 8x320x4096
    float* ssBuf  = ws + off; off += (size_t)BB * NN;
    float* skey   = ws + off; off += (size_t)BB * NN;
    int* siBuf    = (int*)(ws + off); off += (size_t)BB * NN;
    int* sposBuf  = (int*)(ws + off); off += (size_t)BB * NN;
    int* idxArr   = (int*)(ws + off); off += (size_t)BB * KK;

    // --- prep: pad weights/biases, templates, masked index list ---
    pad_w_kernel<<<(M2P * CC + 255) / 256, 256, 0, stream>>>(W2, W2p, VV + 2, CC, M2P, CC);
    pad_w_kernel<<<(M3P * M2P + 255) / 256, 256, 0, stream>>>(W3, W3p, VV + 2, VV + 2, M3P, M2P);
    pad_vec_kernel<<<(M2P + 255) / 256, 256, 0, stream>>>(b2, b2p, VV + 2, M2P);
    pad_vec_kernel<<<(M3P + 255) / 256, 256, 0, stream>>>(b3, b3p, VV + 2, M3P);
    gen_templates_kernel<<<2, 256, 0, stream>>>(tmpl);
    build_idx_kernel<<<BB, 32, 0, stream>>>(index_mask, fp2_inds, idxArr);

    // --- layer 1: Y1 = W1 @ seed_features + b1 (BN applied on read by L2) ---
    {
        dim3 grid(NN / 128, CC / 64, BB);
        gemm_wmma_kernel<false, 0><<<grid, 256, 0, stream>>>(W1, seed_features, b1,
            nullptr, nullptr, Y1, nullptr, nullptr, CC, CC, CC);
    }
    stats_kernel<<<CC, 256, 0, stream>>>(Y1, g1, beta1, scale1, shift1, CC, CC);

    // --- layer 2: Y2 = W2p @ relu(bn(Y1)) + b2p ---
    {
        dim3 grid(NN / 128, M2P / 64, BB);
        gemm_wmma_kernel<true, 0><<<grid, 256, 0, stream>>>(W2p, Y1, b2p,
            scale1, shift1, Y2, nullptr, nullptr, M2P, CC, VV + 2);
    }
    stats_kernel<<<M2P, 256, 0, stream>>>(Y2, g2, beta2, scale2, shift2, M2P, VV + 2);

    // --- layer 3: split-write objectness + transposed view_score ---
    {
        dim3 grid(NN / 128, M3P / 64, BB);
        gemm_wmma_kernel<true, 1><<<grid, 256, 0, stream>>>(W3p, Y2, b3p,
            scale2, shift2, nullptr, out_obj, out_view, M3P, M2P, VV + 2);
    }

    // --- top-2 per masked seed -> ss/si ---
    top2_kernel<<<(BB * KK) / 8, 256, 0, stream>>>(out_view, idxArr, ssBuf, siBuf);

    // --- per-batch full descending sort (== top_k(ss, N)) ---
    sort_kernel<<<BB, 512, 0, stream>>>(ssBuf, skey, sposBuf);

    // --- final gathers + rotation matrices ---
    finalize_kernel<<<(BB * NN + 255) / 256, 256, 0, stream>>>(
        skey, sposBuf, siBuf, idxArr, fp2_xyz, tmpl,
        out_tvs, out_tvi, out_fin, out_fxyz, out_vxyz, out_vrot);
}